// GATBlock_31147102831290
// MI455X (gfx1250) — compile-verified
//
#include <hip/hip_runtime.h>
#include <math.h>

#define Nn 50000
#define Ee 800000
#define Dd 128
#define Hh 4
#define TOTE (Ee + Nn)
#define NEG_SLOPE_F 0.2f
#define EPS_F 1e-5f

typedef float v2f __attribute__((ext_vector_type(2)));
typedef float v8f __attribute__((ext_vector_type(8)));
typedef unsigned u32x4 __attribute__((ext_vector_type(4)));
typedef int i32x4 __attribute__((ext_vector_type(4)));
typedef int i32x8 __attribute__((ext_vector_type(8)));

#if defined(__has_builtin)
#if __has_builtin(__builtin_amdgcn_tensor_load_to_lds) && __has_builtin(__builtin_amdgcn_s_wait_tensorcnt)
#define USE_TDM 1
#endif
#endif
#ifndef USE_TDM
#define USE_TDM 0
#endif

__device__ __forceinline__ float wsum32(float v) {
#pragma unroll
  for (int m = 16; m >= 1; m >>= 1) v += __shfl_xor(v, m, 32);
  return v;
}
__device__ __forceinline__ float wsum16(float v) {
#pragma unroll
  for (int m = 8; m >= 1; m >>= 1) v += __shfl_xor(v, m, 16);
  return v;
}
// order-preserving float->uint map for atomic segment-max
__device__ __forceinline__ unsigned flip_f(float f) {
  unsigned u = __float_as_uint(f);
  return u ^ ((u & 0x80000000u) ? 0xFFFFFFFFu : 0x80000000u);
}
__device__ __forceinline__ float unflip_f(unsigned v) {
  unsigned u = (v & 0x80000000u) ? (v ^ 0x80000000u) : ~v;
  return __uint_as_float(u);
}
__device__ __forceinline__ float gelu_exact(float x) {
  return 0.5f * x * (1.0f + erff(x * 0.70710678118654752f));
}

#if USE_TDM
// Issue a TDM DMA of a (rows x 128) f32 tile starting at `gbase` (row-major,
// row stride 128 floats, tensor has tensor_rows total rows) into LDS at
// lds_byte. D# layout per CDNA5 ISA. 2D tensor ->
// remaining descriptor groups zero-filled (6-arg toolchain variant).
__device__ __forceinline__ void tdm_load_tile_f32_128(const float* gbase,
                                                      unsigned lds_byte,
                                                      unsigned tile_rows,
                                                      unsigned tensor_rows) {
  unsigned long long ga = (unsigned long long)(const void*)gbase;
  u32x4 g0;
  g0[0] = 1u;                                   // count=1, user descriptor
  g0[1] = lds_byte;                             // lds_addr
  g0[2] = (unsigned)ga;                         // global_addr[31:0]
  g0[3] = ((unsigned)(ga >> 32) & 0x01FFFFFFu)  // global_addr[56:32]
          | 0x80000000u;                        // type=2 ("image")
  i32x8 g1;
  g1[0] = (int)(2u << 16);                      // data_size=2 (4 bytes)
  g1[1] = (int)(128u << 16);                    // tensor_dim0[15:0]=128
  g1[2] = (int)((tensor_rows & 0xFFFFu) << 16); // dim0 hi=0 | tensor_dim1 lo
  g1[3] = (int)((128u << 16) | (tensor_rows >> 16)); // tensor_dim1 hi | tile_dim0=128
  g1[4] = (int)(tile_rows & 0xFFFFu);           // tile_dim1 | tile_dim2=0
  g1[5] = 128;                                  // tensor_dim0_stride lo32 = 128
  g1[6] = 0;                                    // stride hi | dim1_stride lo
  g1[7] = 0;
  i32x4 gz4 = {0, 0, 0, 0};
  i32x8 gz8 = {0, 0, 0, 0, 0, 0, 0, 0};
  __builtin_amdgcn_tensor_load_to_lds(g0, g1, gz4, gz4, gz8, 0);
  __builtin_amdgcn_s_wait_tensorcnt(0);
}
#endif

// ---------------- Kernel 1: LN1 + residual/bias init + smax/denom init ----
__global__ void k_ln1_init(const float* __restrict__ x, const float* __restrict__ ln1_w,
                           const float* __restrict__ gat_bias,
                           float* __restrict__ h, float* __restrict__ x1,
                           unsigned* __restrict__ smax, float* __restrict__ denom) {
  int lane = threadIdx.x & 31;
  int wid = threadIdx.x >> 5;
  int row = blockIdx.x * 8 + wid;
  if (row >= Nn) return;
  float4 v = ((const float4*)(x + (size_t)row * Dd))[lane];
  float mean = wsum32(v.x + v.y + v.z + v.w) * (1.0f / Dd);
  float dx = v.x - mean, dy = v.y - mean, dz = v.z - mean, dw = v.w - mean;
  float var = wsum32(dx * dx + dy * dy + dz * dz + dw * dw) * (1.0f / Dd);
  float rstd = rsqrtf(var + EPS_F);
  float4 w = ((const float4*)ln1_w)[lane];
  float4 hb;
  hb.x = dx * rstd * w.x; hb.y = dy * rstd * w.y;
  hb.z = dz * rstd * w.z; hb.w = dw * rstd * w.w;
  ((float4*)(h + (size_t)row * Dd))[lane] = hb;
  float4 gb = ((const float4*)gat_bias)[lane];
  float4 x1v;
  x1v.x = v.x + gb.x; x1v.y = v.y + gb.y; x1v.z = v.z + gb.z; x1v.w = v.w + gb.w;
  ((float4*)(x1 + (size_t)row * Dd))[lane] = x1v;
  if (lane < Hh) {
    smax[row * Hh + lane] = 0u;          // < flip(any finite score)
    denom[row * Hh + lane] = 0.0f;
  }
}

// ---------------- Kernel 2: xl = h@Wl+bl, xr = h@Wr+br via f32 WMMA -------
__global__ void __launch_bounds__(256) k_gemm_xlxr(
    const float* __restrict__ h,
    const float* __restrict__ Wl, const float* __restrict__ bl,
    const float* __restrict__ Wr, const float* __restrict__ br,
    float* __restrict__ xl, float* __restrict__ xr) {
  __shared__ float Ah[16][Dd];
  int tid = threadIdx.x;
  int row0 = blockIdx.x * 16;
#if USE_TDM
  if (tid < 32) {
    tdm_load_tile_f32_128(h + (size_t)row0 * Dd,
                          (unsigned)(size_t)(&Ah[0][0]), 16u, (unsigned)Nn);
  }
#else
  {
    int r = tid >> 4;
    int c0 = (tid & 15) * 8;
    const float4* src = (const float4*)(h + (size_t)(row0 + r) * Dd + c0);
    *((float4*)&Ah[r][c0]) = src[0];
    *((float4*)&Ah[r][c0 + 4]) = src[1];
  }
#endif
  __syncthreads();
  int lane = tid & 31, wid = tid >> 5;
  int ml = lane & 15;     // M row within fragment
  int kh = lane >> 4;     // lane half selects K sub-pair
  int col = wid * 16 + ml;
  v8f accL = {};
  v8f accR = {};
#pragma unroll 4
  for (int k = 0; k < Dd; k += 4) {
    v2f a;  // A 16x4 f32: lanes0-15 K=k,k+1 ; lanes16-31 K=k+2,k+3
    a.x = Ah[ml][k + 2 * kh];
    a.y = Ah[ml][k + 2 * kh + 1];
    v2f bL, bR;  // B 4x16 f32: rows striped across lane halves
    bL.x = Wl[(size_t)(k + kh) * Dd + col];
    bL.y = Wl[(size_t)(k + 2 + kh) * Dd + col];
    bR.x = Wr[(size_t)(k + kh) * Dd + col];
    bR.y = Wr[(size_t)(k + 2 + kh) * Dd + col];
    accL = __builtin_amdgcn_wmma_f32_16x16x4_f32(false, a, false, bL, (short)0, accL, false, false);
    accR = __builtin_amdgcn_wmma_f32_16x16x4_f32(false, a, false, bR, (short)0, accR, false, false);
  }
  float blb = bl[col], brb = br[col];
#pragma unroll
  for (int v = 0; v < 8; ++v) {
    int r = row0 + v + kh * 8;  // C VGPR v: lanes0-15 M=v ; lanes16-31 M=v+8
    xl[(size_t)r * Dd + col] = accL[v] + blb;
    xr[(size_t)r * Dd + col] = accR[v] + brb;
  }
}

// ---------------- Kernel 3: edge scores + atomic segment max --------------
__global__ void k_edge_score(const int* __restrict__ ei,
                             const float* __restrict__ xl, const float* __restrict__ xr,
                             const float* __restrict__ att, float* __restrict__ score,
                             unsigned* __restrict__ smax) {
  int lane = threadIdx.x & 31;
  int wid = threadIdx.x >> 5;
  int e = blockIdx.x * 8 + wid;
  if (e >= TOTE) return;
  int src, dst;
  if (e < Ee) { src = ei[e]; dst = ei[Ee + e]; } else { src = dst = e - Ee; }
  const float* xls = xl + (size_t)src * Dd;
  const float* xrd = xr + (size_t)dst * Dd;
  __builtin_prefetch(xls, 0, 3);  // global_prefetch: warm line for the agg pass
  float sh[4];
#pragma unroll
  for (int hh = 0; hh < 4; ++hh) {
    int idx = hh * 32 + lane;
    float v = xls[idx] + xrd[idx];
    v = v > 0.0f ? v : NEG_SLOPE_F * v;
    sh[hh] = att[idx] * v;
  }
#pragma unroll
  for (int hh = 0; hh < 4; ++hh) sh[hh] = wsum32(sh[hh]);
  if (lane == 0) {
    float4 sc; sc.x = sh[0]; sc.y = sh[1]; sc.z = sh[2]; sc.w = sh[3];
    ((float4*)score)[e] = sc;
    atomicMax(&smax[dst * 4 + 0], flip_f(sh[0]));
    atomicMax(&smax[dst * 4 + 1], flip_f(sh[1]));
    atomicMax(&smax[dst * 4 + 2], flip_f(sh[2]));
    atomicMax(&smax[dst * 4 + 3], flip_f(sh[3]));
  }
}

// ---------------- Kernel 4: ex = exp(score - smax[dst]); denom += ex ------
__global__ void k_edge_ex(const int* __restrict__ ei, float* __restrict__ score,
                          const unsigned* __restrict__ smax, float* __restrict__ denom) {
  int e = blockIdx.x * blockDim.x + threadIdx.x;
  if (e >= TOTE) return;
  int dst = (e < Ee) ? ei[Ee + e] : (e - Ee);
  float4 sc = ((const float4*)score)[e];
  float4 ex;
  ex.x = __expf(sc.x - unflip_f(smax[dst * 4 + 0]));
  ex.y = __expf(sc.y - unflip_f(smax[dst * 4 + 1]));
  ex.z = __expf(sc.z - unflip_f(smax[dst * 4 + 2]));
  ex.w = __expf(sc.w - unflip_f(smax[dst * 4 + 3]));
  ((float4*)score)[e] = ex;  // in-place: score buffer becomes ex buffer
  __hip_atomic_fetch_add(&denom[dst * 4 + 0], ex.x, __ATOMIC_RELAXED, __HIP_MEMORY_SCOPE_AGENT);
  __hip_atomic_fetch_add(&denom[dst * 4 + 1], ex.y, __ATOMIC_RELAXED, __HIP_MEMORY_SCOPE_AGENT);
  __hip_atomic_fetch_add(&denom[dst * 4 + 2], ex.z, __ATOMIC_RELAXED, __HIP_MEMORY_SCOPE_AGENT);
  __hip_atomic_fetch_add(&denom[dst * 4 + 3], ex.w, __ATOMIC_RELAXED, __HIP_MEMORY_SCOPE_AGENT);
}

// ---------------- Kernel 5: x1[dst] += alpha * xl[src] --------------------
__global__ void k_edge_agg(const int* __restrict__ ei, const float* __restrict__ xl,
                           const float* __restrict__ ex, const float* __restrict__ denom,
                           float* __restrict__ x1) {
  int lane = threadIdx.x & 31;
  int wid = threadIdx.x >> 5;
  int e = blockIdx.x * 8 + wid;
  if (e >= TOTE) return;
  int src, dst;
  if (e < Ee) { src = ei[e]; dst = ei[Ee + e]; } else { src = dst = e - Ee; }
  float4 exv = ((const float4*)ex)[e];
  float4 dnv = *((const float4*)(denom + (size_t)dst * 4));
  float al[4] = {exv.x / dnv.x, exv.y / dnv.y, exv.z / dnv.z, exv.w / dnv.w};
  const float* xls = xl + (size_t)src * Dd;
  float* xd = x1 + (size_t)dst * Dd;
#pragma unroll
  for (int hh = 0; hh < 4; ++hh) {
    int idx = hh * 32 + lane;
    __hip_atomic_fetch_add(&xd[idx], al[hh] * xls[idx], __ATOMIC_RELAXED, __HIP_MEMORY_SCOPE_AGENT);
  }
}

// ---------------- Kernel 6: fused LN2 + FFN (WMMA) + residual -------------
__global__ void __launch_bounds__(256) k_ffn(const float* __restrict__ x1,
                                             const float* __restrict__ ln2_w,
                                             const float* __restrict__ Wfc,
                                             const float* __restrict__ Wproj,
                                             float* __restrict__ out) {
  __shared__ float xs[16][Dd];        // residual tile (8 KB)
  __shared__ float hs[16][Dd];        // LN2 output (8 KB)
  __shared__ float gs[16][4 * Dd];    // gelu(h2@Wfc) (32 KB)
  int tid = threadIdx.x;
  int row0 = blockIdx.x * 16;
#if USE_TDM
  if (tid < 32) {
    tdm_load_tile_f32_128(x1 + (size_t)row0 * Dd,
                          (unsigned)(size_t)(&xs[0][0]), 16u, (unsigned)Nn);
  }
  __syncthreads();
  {
    int r = tid >> 4;
    int c0 = (tid & 15) * 8;
    float v[8];
    float s = 0.0f;
#pragma unroll
    for (int i = 0; i < 8; ++i) { v[i] = xs[r][c0 + i]; s += v[i]; }
    float mean = wsum16(s) * (1.0f / Dd);
    float vs = 0.0f;
#pragma unroll
    for (int i = 0; i < 8; ++i) { float d = v[i] - mean; vs += d * d; }
    float rstd = rsqrtf(wsum16(vs) * (1.0f / Dd) + EPS_F);
#pragma unroll
    for (int i = 0; i < 8; ++i) hs[r][c0 + i] = (v[i] - mean) * rstd * ln2_w[c0 + i];
  }
#else
  {
    int r = tid >> 4;
    int c0 = (tid & 15) * 8;
    const float* src = x1 + (size_t)(row0 + r) * Dd + c0;
    float v[8];
    float s = 0.0f;
#pragma unroll
    for (int i = 0; i < 8; ++i) { v[i] = src[i]; s += v[i]; }
#pragma unroll
    for (int i = 0; i < 8; ++i) xs[r][c0 + i] = v[i];
    float mean = wsum16(s) * (1.0f / Dd);
    float vs = 0.0f;
#pragma unroll
    for (int i = 0; i < 8; ++i) { float d = v[i] - mean; vs += d * d; }
    float rstd = rsqrtf(wsum16(vs) * (1.0f / Dd) + EPS_F);
#pragma unroll
    for (int i = 0; i < 8; ++i) hs[r][c0 + i] = (v[i] - mean) * rstd * ln2_w[c0 + i];
  }
#endif
  __syncthreads();
  int lane = tid & 31, wid = tid >> 5;
  int ml = lane & 15, kh = lane >> 4;

  // GEMM1: 16x128 @ 128x512 ; wave owns 64 cols (4 WMMA tiles)
  v8f acc[4] = {};
  for (int k = 0; k < Dd; k += 4) {
    v2f a;
    a.x = hs[ml][k + 2 * kh];
    a.y = hs[ml][k + 2 * kh + 1];
#pragma unroll
    for (int j = 0; j < 4; ++j) {
      int coln = wid * 64 + j * 16 + ml;
      v2f b;
      b.x = Wfc[(size_t)(k + kh) * (4 * Dd) + coln];
      b.y = Wfc[(size_t)(k + 2 + kh) * (4 * Dd) + coln];
      acc[j] = __builtin_amdgcn_wmma_f32_16x16x4_f32(false, a, false, b, (short)0, acc[j], false, false);
    }
  }
#pragma unroll
  for (int j = 0; j < 4; ++j) {
    int coln = wid * 64 + j * 16 + ml;
#pragma unroll
    for (int v = 0; v < 8; ++v) gs[v + kh * 8][coln] = gelu_exact(acc[j][v]);
  }
  __syncthreads();

  // GEMM2: 16x512 @ 512x128 ; wave owns 16 cols
  int col = wid * 16 + ml;
  v8f acc2 = {};
  for (int k = 0; k < 4 * Dd; k += 4) {
    v2f a;
    a.x = gs[ml][k + 2 * kh];
    a.y = gs[ml][k + 2 * kh + 1];
    v2f b;
    b.x = Wproj[(size_t)(k + kh) * Dd + col];
    b.y = Wproj[(size_t)(k + 2 + kh) * Dd + col];
    acc2 = __builtin_amdgcn_wmma_f32_16x16x4_f32(false, a, false, b, (short)0, acc2, false, false);
  }
#pragma unroll
  for (int v = 0; v < 8; ++v) {
    int r = v + kh * 8;
    out[(size_t)(row0 + r) * Dd + col] = xs[r][col] + acc2[v];
  }
}

// ---------------- Host-side launch ----------------------------------------
extern "C" void kernel_launch(void* const* d_in, const int* in_sizes, int n_in,
                              void* d_out, int out_size, void* d_ws, size_t ws_size,
                              hipStream_t stream) {
  (void)in_sizes; (void)n_in; (void)out_size; (void)ws_size;
  const float* x        = (const float*)d_in[0];
  const int*   ei       = (const int*)d_in[1];
  const float* ln1_w    = (const float*)d_in[2];
  const float* Wl       = (const float*)d_in[3];
  const float* bl       = (const float*)d_in[4];
  const float* Wr       = (const float*)d_in[5];
  const float* br       = (const float*)d_in[6];
  const float* att      = (const float*)d_in[7];
  const float* gat_bias = (const float*)d_in[8];
  const float* ln2_w    = (const float*)d_in[9];
  const float* Wfc      = (const float*)d_in[10];
  const float* Wproj    = (const float*)d_in[11];
  float* out = (float*)d_out;

  float* ws    = (float*)d_ws;
  float* h     = ws;                                   // N*D
  float* xl    = ws + (size_t)Nn * Dd;                 // N*D
  float* xr    = ws + 2 * (size_t)Nn * Dd;             // N*D
  float* score = ws + 3 * (size_t)Nn * Dd;             // TOTE*H (becomes ex)
  unsigned* smax = (unsigned*)(score + (size_t)TOTE * Hh);  // N*H
  float* denom = (float*)(smax + (size_t)Nn * Hh);     // N*H

  k_ln1_init<<<(Nn + 7) / 8, 256, 0, stream>>>(x, ln1_w, gat_bias, h, out, smax, denom);
  k_gemm_xlxr<<<Nn / 16, 256, 0, stream>>>(h, Wl, bl, Wr, br, xl, xr);
  k_edge_score<<<(TOTE + 7) / 8, 256, 0, stream>>>(ei, xl, xr, att, score, smax);
  k_edge_ex<<<(TOTE + 255) / 256, 256, 0, stream>>>(ei, score, smax, denom);
  k_edge_agg<<<(TOTE + 7) / 8, 256, 0, stream>>>(ei, xl, score, denom, out);
  k_ffn<<<Nn / 16, 256, 0, stream>>>(out, ln2_w, Wfc, Wproj, out);
}